// CrossLayer_33346126086777
// MI455X (gfx1250) — compile-verified
//
#include <hip/hip_runtime.h>
#include <hip/hip_bf16.h>

typedef __attribute__((ext_vector_type(2))) float v2f;
typedef __attribute__((ext_vector_type(8))) float v8f;

#define BB 256
#define FF 39
#define FP 48           // padded feature count (3 tiles of 16)
#define EE 16
#define II 741          // 39*38/2
#define NPAIRS (BB * II)
#define BN_EPS 1e-5f

// ---------------- Kernel A: per-(b,f) sums, transposed [f][b] with zero padding to FP rows ----
__global__ void stats_pre(const float* __restrict__ xi, const float* __restrict__ xj,
                          float* __restrict__ Si, float* __restrict__ Qi,
                          float* __restrict__ Sj, float* __restrict__ Qj) {
    int f = blockIdx.x;    // 0..47
    int b = threadIdx.x;   // 0..255
    float si = 0.f, qi = 0.f, sj = 0.f, qj = 0.f;
    if (f < FF) {
        const float4* pi = (const float4*)(xi + (b * FF + f) * EE);
        const float4* pj = (const float4*)(xj + (b * FF + f) * EE);
#pragma unroll
        for (int k = 0; k < 4; ++k) {
            float4 a = pi[k];
            si += a.x + a.y + a.z + a.w;
            qi += a.x * a.x + a.y * a.y + a.z * a.z + a.w * a.w;
            float4 c = pj[k];
            sj += c.x + c.y + c.z + c.w;
            qj += c.x * c.x + c.y * c.y + c.z * c.z + c.w * c.w;
        }
    }
    Si[f * BB + b] = si; Qi[f * BB + b] = qi;   // rows 39..47 are zeros -> no guards in GEMM
    Sj[f * BB + b] = sj; Qj[f * BB + b] = qj;
}

// ---------------- Kernel B: 48x48 stats GEMM (K=256) via WMMA f32 16x16x4 ----------------
// mean[f1,f0] = (1/65536) * sum_b Sj[f1,b]*Si[f0,b];  msq likewise from Q.
// Emits fused channel table tab[i] = { f1, f0, scale, shift }.
__global__ void stats_gemm(const float* __restrict__ Si, const float* __restrict__ Qi,
                           const float* __restrict__ Sj, const float* __restrict__ Qj,
                           const float* __restrict__ gamma, const float* __restrict__ beta,
                           float4* __restrict__ tab) {
    int lane = threadIdx.x & 31;
    int wave = threadIdx.x >> 5;          // 0..8  (3x3 tiles of 16x16)
    int tr = wave / 3, tc = wave % 3;
    int half = lane >> 4, idx = lane & 15;
    int m = tr * 16 + idx;                // f1 row fed to A (always < FP, pad rows are zero)
    int n = tc * 16 + idx;                // f0 row fed to B

    const v2f* aS = (const v2f*)(Sj + m * BB);
    const v2f* aQ = (const v2f*)(Qj + m * BB);
    const v2f* bS = (const v2f*)(Si + n * BB);
    const v2f* bQ = (const v2f*)(Qi + n * BB);

    v8f accm = {}; v8f accq = {};
    for (int k0 = 0; k0 < BB; k0 += 4) {
        int kh = (k0 >> 1) + half;        // VGPR0: K=k0+2*half, VGPR1: K=k0+2*half+1
        v2f am = aS[kh], aq = aQ[kh];
        v2f bm = bS[kh], bq = bQ[kh];
        accm = __builtin_amdgcn_wmma_f32_16x16x4_f32(false, am, false, bm,
                                                     (short)0, accm, false, false);
        accq = __builtin_amdgcn_wmma_f32_16x16x4_f32(false, aq, false, bq,
                                                     (short)0, accq, false, false);
    }

    const float invN = 1.0f / 65536.0f;   // 1/(B*E*E)
#pragma unroll
    for (int v = 0; v < 8; ++v) {
        int f1 = tr * 16 + v + 8 * half;  // D layout: M = v + 8*(lane/16)
        int f0 = tc * 16 + idx;           // N = lane%16
        if (f1 < FF && f0 < f1) {
            int i = f1 * (f1 - 1) / 2 + f0;
            float mean = accm[v] * invN;
            float msq  = accq[v] * invN;
            float var  = msq - mean * mean;
            float rinv = rsqrtf(var + BN_EPS);
            float sc = gamma[i] * rinv;
            float sh = beta[i] - mean * sc;
            tab[i] = make_float4(__int_as_float(f1), __int_as_float(f0), sc, sh);
        }
    }
}

// ---------------- Kernel C: streaming output. One wave per (b,i) pair, 8 pairs/wave ----------
// out[b,0,i,e1,e2] = (xj[b,r,e1]*xi[b,c,e2] * scale[i] + shift[i]) * W[e1,e2]
__global__ void main_out(const float* __restrict__ xi, const float* __restrict__ xj,
                         const float* __restrict__ W,
                         const float4* __restrict__ tab,
                         float* __restrict__ out) {
    int lane = threadIdx.x & 31;
    int half = lane >> 4, idx = lane & 15;
    // Force wave-uniform pair indexing -> SGPR math + scalar channel-table loads
    int wave = __builtin_amdgcn_readfirstlane(blockIdx.x * (blockDim.x >> 5) + (threadIdx.x >> 5));

    // Preload this lane's 8 W values matching the C/D layout (M = v + 8*half, N = idx)
    float wreg[8];
#pragma unroll
    for (int v = 0; v < 8; ++v) wreg[v] = W[(v + 8 * half) * EE + idx];

    int p0 = wave * 8;                    // NPAIRS = 23712 waves * 8 exactly
#pragma unroll 1
    for (int q = 0; q < 8; ++q) {
        int p = p0 + q;
        if (p >= NPAIRS) return;          // uniform branch; EXEC all-ones at WMMA
        int b = p / II;
        int i = p - b * II;

        float4 tv = tab[i];               // uniform address -> scalar load
        int r = __float_as_int(tv.x);
        int c = __float_as_int(tv.y);
        float s = tv.z, t = tv.w;

        // Unconditional loads (lanes 16-31 hit the same cacheline), select K=0 column only
        float av = xj[(b * FF + r) * EE + idx];
        float bw = xi[(b * FF + c) * EE + idx];
        v2f a  = { half ? 0.0f : av, 0.0f };
        v2f bv = { half ? 0.0f : bw, 0.0f };

        v8f acc = {};
        acc = __builtin_amdgcn_wmma_f32_16x16x4_f32(false, a, false, bv,
                                                    (short)0, acc, false, false);

        float* op = out + (size_t)p * 256;
#pragma unroll
        for (int v = 0; v < 8; ++v) {
            __builtin_nontemporal_store(fmaf(acc[v], s, t) * wreg[v],
                                        op + (v + 8 * half) * EE + idx);
        }
    }
}

extern "C" void kernel_launch(void* const* d_in, const int* in_sizes, int n_in,
                              void* d_out, int out_size, void* d_ws, size_t ws_size,
                              hipStream_t stream) {
    const float* xi    = (const float*)d_in[0];   // (256,39,16)
    const float* xj    = (const float*)d_in[1];   // (256,39,16)
    const float* W     = (const float*)d_in[2];   // (1,16,16)
    const float* gamma = (const float*)d_in[3];   // (741)
    const float* beta  = (const float*)d_in[4];   // (741)
    float* out = (float*)d_out;                   // (256,1,741,16,16)

    // Workspace layout (floats): 4 transposed padded sum arrays + fused channel table
    float*  ws  = (float*)d_ws;
    float*  Si  = ws;                    // 48*256
    float*  Qi  = Si + FP * BB;          // 48*256
    float*  Sj  = Qi + FP * BB;          // 48*256
    float*  Qj  = Sj + FP * BB;          // 48*256
    float4* tab = (float4*)(Qj + FP * BB); // 741 * 16B   (~208 KB total)

    // A: per-row sums (transposed + zero-padded to 48 feature rows)
    stats_pre<<<FP, BB, 0, stream>>>(xi, xj, Si, Qi, Sj, Qj);

    // B: 48x48 x K=256 stats GEMM on WMMA; emits fused {r,c,scale,shift} table
    stats_gemm<<<1, 288, 0, stream>>>(Si, Qi, Sj, Qj, gamma, beta, tab);

    // C: 189,696 rank-1 WMMA outer products + non-temporal streaming store (194 MB)
    // 23712 waves total, 8 waves per 256-thread block -> 2964 blocks (exact)
    main_out<<<NPAIRS / (8 * 8), 256, 0, stream>>>(xi, xj, W, tab, out);
}